// PerResidueLDDAllAtom_90228672954830
// MI455X (gfx1250) — compile-verified
//
#include <hip/hip_runtime.h>

#define B_     2
#define N_RES  2048
#define N_ATOM 16384
#define C_S    384
#define C_OUT  50

// LDS row stride (floats): 384 + 4 so that 16 rows at the same column hit
// 16 distinct banks (stride % 64 == 4 -> row*4 mod 64 all distinct for 16 rows).
#define LDS_STRIDE 388

typedef float v2f __attribute__((ext_vector_type(2)));
typedef float v8f __attribute__((ext_vector_type(8)));
typedef float f4  __attribute__((ext_vector_type(4)));

// ---------------------------------------------------------------------------
// Kernel 1: recover per-atom residue index from the one-hot matrix.
// One thread scans 4 consecutive residues of one (b, atom) row.
// The 268 MB one-hot is streamed exactly once with nontemporal loads.
// ---------------------------------------------------------------------------
__global__ void __launch_bounds__(256)
onehot_to_index_kernel(const float* __restrict__ onehot, int* __restrict__ idx) {
    unsigned t = blockIdx.x * 256u + threadIdx.x;
    unsigned e = t * 4u;                      // flat element into [B*N_ATOM, N_RES]
    const f4 v = __builtin_nontemporal_load((const f4*)(onehot + e));
    unsigned row = e >> 11;                   // / N_RES (=2048)
    unsigned r   = e & 2047u;                 // % N_RES
    if (v.x != 0.0f) idx[row] = (int)r;
    if (v.y != 0.0f) idx[row] = (int)(r + 1u);
    if (v.z != 0.0f) idx[row] = (int)(r + 2u);
    if (v.w != 0.0f) idx[row] = (int)(r + 3u);
}

// ---------------------------------------------------------------------------
// Kernel 2: gather s rows for a 16-atom tile into LDS, then f32 WMMA GEMM
// against W^T (N = 50 padded to 64, one 16-wide N-slice per wave).
//
// V_WMMA_F32_16X16X4_F32 operand layout (wave32):
//   A (16x4, M x K): lane l holds M = l%16; VGPR0/1 hold K = 2*(l/16) + {0,1}
//   B (4x16, K x N): lane l holds N = l%16; VGPR0/1 hold K = 2*(l/16) + {0,1}
//   C/D (16x16):     lane l holds N = l%16; VGPR r holds M = r + 8*(l/16)
//
// Padded columns n in [50,64) load W row 0 (valid memory); each D column
// depends only on its own B column, and those columns are never stored,
// so no zeroing / EXEC predication is needed in the hot loop.
// ---------------------------------------------------------------------------
__global__ void __launch_bounds__(128)
gather_gemm_wmma_kernel(const float* __restrict__ s,
                        const float* __restrict__ W,
                        const float* __restrict__ bias,
                        const int*   __restrict__ idx,
                        float* __restrict__ out) {
    __shared__ float sA[16 * LDS_STRIDE];

    const int tile  = blockIdx.x;                 // 0 .. B*N_ATOM/16 - 1
    const int b     = tile / (N_ATOM / 16);
    const int atom0 = (tile % (N_ATOM / 16)) * 16;
    const int lane  = threadIdx.x & 31;
    const int wave  = threadIdx.x >> 5;           // 0..3 -> output slice

    // ---- Stage 16 gathered rows of s into LDS (float2 granularity) ----
    for (int i = threadIdx.x; i < 16 * (C_S / 2); i += 128) {
        const int row  = i / (C_S / 2);
        const int col2 = i % (C_S / 2);
        const int res  = idx[b * N_ATOM + atom0 + row];
        const float2 v = *(const float2*)(s + ((long long)b * N_RES + res) * C_S + col2 * 2);
        *(float2*)(&sA[row * LDS_STRIDE + col2 * 2]) = v;
    }
    __syncthreads();

    const int  m    = lane & 15;                  // A-matrix row for this lane
    const int  koff = (lane >> 4) * 2;            // K sub-offset per half-wave
    const int  n    = wave * 16 + (lane & 15);    // output column
    const bool nval = (n < C_OUT);
    // Clamp invalid columns to W row 0: valid memory, never stored.
    const float* wrow = W + (long long)(nval ? n : 0) * C_S;
    const float* aptr = &sA[m * LDS_STRIDE + koff];

    v8f c = {0.f, 0.f, 0.f, 0.f, 0.f, 0.f, 0.f, 0.f};

    for (int k = 0; k < C_S; k += 4) {
        v2f a, bb;
        a.x  = aptr[k];
        a.y  = aptr[k + 1];
        bb.x = wrow[k + koff];                    // unconditional: no EXEC churn
        bb.y = wrow[k + koff + 1];
        // D = A x B + C, full fp32 (matches reference einsum precision class)
        c = __builtin_amdgcn_wmma_f32_16x16x4_f32(
                /*neg_a=*/false, a, /*neg_b=*/false, bb,
                /*c_mod=*/(short)0, c, /*reuse_a=*/false, /*reuse_b=*/false);
    }

    // ---- Epilogue: bias add + store (only the 50 valid columns) ----
    if (nval) {
        const float bv    = bias[n];
        const int   mhalf = (lane >> 4) * 8;
        #pragma unroll
        for (int r = 0; r < 8; ++r) {
            const int mo = r + mhalf;
            out[(((long long)b * N_ATOM) + atom0 + mo) * C_OUT + n] = c[r] + bv;
        }
    }
}

// ---------------------------------------------------------------------------
extern "C" void kernel_launch(void* const* d_in, const int* in_sizes, int n_in,
                              void* d_out, int out_size, void* d_ws, size_t ws_size,
                              hipStream_t stream) {
    const float* s      = (const float*)d_in[0];  // [B, N_RES, C_S]
    const float* onehot = (const float*)d_in[1];  // [B, N_ATOM, N_RES]
    const float* W      = (const float*)d_in[2];  // [C_OUT, C_S]
    const float* bias   = (const float*)d_in[3];  // [C_OUT]
    float* out          = (float*)d_out;          // [B, N_ATOM, C_OUT]
    int*   idx          = (int*)d_ws;             // [B * N_ATOM] = 128 KB scratch

    // Kernel 1: 67,108,864 one-hot elements / 4 per thread / 256 per block
    const unsigned total   = (unsigned)B_ * N_ATOM * N_RES;
    const unsigned blocks1 = total / 4u / 256u;   // = 65536, exact cover
    onehot_to_index_kernel<<<blocks1, 256, 0, stream>>>(onehot, idx);

    // Kernel 2: one block per 16-atom tile, 4 waves cover the 50 outputs
    const unsigned blocks2 = (unsigned)B_ * (N_ATOM / 16);  // = 2048
    gather_gemm_wmma_kernel<<<blocks2, 128, 0, stream>>>(s, W, bias, idx, out);
}